// PATCHY_LAYER_RETURNFULLSEQ_43997644980705
// MI455X (gfx1250) — compile-verified
//
#include <hip/hip_runtime.h>

// PATCHY_LAYER_RETURNFULLSEQ for MI455X (gfx1250, wave32).
//
// out[b,t,0:400]   = leakyrelu(gatherdot(x, idx_main[t/8, p], W_main) + b_main, 0.3)
// out[b,t,400:460] = leakyrelu(gatherdot(x, idx_top[t, p],    W_top ) + b_top , 0.3)
//
// One wave per (s,p) patch task. The K=128 reduction across the 16-batch
// dimension runs on V_WMMA_F32_16X16X4_F32:
//   A[m=batch, k] = gathered x values (L2-resident, zero-redundancy gathers)
//   B[k, n]       = W[patch, k] broadcast across all 16 columns
//   D[m, n]       = dot result, identical across n -> columns double as the
//                   x8 time-repeat replicas for the main layer's stores.
//
// Roofline: ~147 MB of HBM traffic (idx+W streams 115 MB NT, out 30 MB NT,
// x 2 MB resident) -> ~6.3 us floor at 23.3 TB/s; FLOPs are negligible even
// with the 16x WMMA column redundancy.

typedef __attribute__((ext_vector_type(2))) float v2f;
typedef __attribute__((ext_vector_type(8))) float v8f;

#define B_SZ   16
#define T_SZ   1024
#define TC_SZ  32768          // T*C
#define K_SZ   128
#define NP     400            // NB_PATCHES (main)
#define PTOP   60             // ontop patches
#define OUTC   460            // NP + PTOP
#define NMAIN  (128 * 400)    // S_MAIN * NP   = 51200 tasks
#define NTOP   (1024 * 60)    // T * PTOP      = 61440 tasks
#define NTASK  (NMAIN + NTOP) // 112640
#define WPB    8              // waves per block (256 threads)

__global__ __launch_bounds__(256)
void patchy_wmma_kernel(const float* __restrict__ x,
                        const float* __restrict__ Wm,
                        const float* __restrict__ bm,
                        const float* __restrict__ Wt,
                        const float* __restrict__ bt,
                        const int*   __restrict__ Im,
                        const int*   __restrict__ It,
                        float*       __restrict__ out)
{
    const int lane = threadIdx.x & 31;
    const int wave = threadIdx.x >> 5;
    const int task = blockIdx.x * WPB + wave;     // grid sized exactly: no guard

    const int b      = lane & 15;                 // WMMA A-matrix M index = batch
    const int hi     = lane >> 4;                 // lane half -> K pair {0,1} / {2,3}
    const int rowoff = b * TC_SZ;                 // 32-bit element offset of batch row

    // ---- wave-uniform task decode (EXEC stays all-1s through the WMMA loop) ----
    const bool is_main = (task < NMAIN);
    int s, p;
    const unsigned long long* __restrict__ ip2;   // idx pairs   (b64 NT stream)
    const unsigned long long* __restrict__ wp2;   // weight pairs(b64 NT stream)
    float bias;
    if (is_main) {
        s    = task / NP;                         // strided time step
        p    = task - s * NP;                     // -> output channel p
        ip2  = (const unsigned long long*)(Im + (size_t)task * K_SZ);
        wp2  = (const unsigned long long*)(Wm + (size_t)task * K_SZ);
        bias = bm[task];
    } else {
        const int tt = task - NMAIN;
        s    = tt / PTOP;                         // full-rate time step t
        p    = tt - s * PTOP;                     // -> output channel 400 + p
        ip2  = (const unsigned long long*)(It + (size_t)tt * K_SZ);
        wp2  = (const unsigned long long*)(Wt + (size_t)tt * K_SZ);
        bias = bt[tt];
    }

    // Lane needs k = {4c + 2*hi, 4c + 2*hi + 1} per chunk c  ->  pair index 2c + hi.
    v8f acc = {};
    #pragma unroll 8
    for (int c = 0; c < 32; ++c) {
        const int pr = 2 * c + hi;
        const unsigned long long iv = __builtin_nontemporal_load(ip2 + pr);
        const unsigned long long wv = __builtin_nontemporal_load(wp2 + pr);
        const int i0 = (int)(unsigned)(iv & 0xffffffffu);
        const int i1 = (int)(unsigned)(iv >> 32);
        v2f av;                                    // gathered A tile (RT: keep x hot)
        av.x = x[rowoff + i0];
        av.y = x[rowoff + i1];
        v2f bv;                                    // weight broadcast to all 16 cols
        bv.x = __builtin_bit_cast(float, (unsigned)(wv & 0xffffffffu));
        bv.y = __builtin_bit_cast(float, (unsigned)(wv >> 32));
        acc = __builtin_amdgcn_wmma_f32_16x16x4_f32(
                  /*neg_a=*/false, av, /*neg_b=*/false, bv,
                  /*c_mod=*/(short)0, acc, /*reuse_a=*/false, /*reuse_b=*/false);
    }

    // bias + leaky_relu(0.3) == max(v, 0.3v) since slope < 1  (mul+max, 2 VALU)
    float res[8];
    #pragma unroll
    for (int r = 0; r < 8; ++r) {
        const float v = acc[r] + bias;
        res[r] = fmaxf(v, 0.3f * v);
    }

    const int n = lane & 15;                      // D column (all columns equal)
    if (is_main) {
        // time-repeat x8: column n (n<8) writes replica t = s*8 + n
        if (n < 8) {
            const int t = s * 8 + n;
            #pragma unroll
            for (int r = 0; r < 8; ++r) {
                const int bb  = r + 8 * hi;       // batch = D row
                const int off = (bb * T_SZ + t) * OUTC + p;
                __builtin_nontemporal_store(res[r], out + off);
            }
        }
    } else {
        // single time step: column 0 only (lanes 0 and 16 cover b=0..7 / 8..15)
        if (n == 0) {
            #pragma unroll
            for (int r = 0; r < 8; ++r) {
                const int bb  = r + 8 * hi;
                const int off = (bb * T_SZ + s) * OUTC + (NP + p);
                __builtin_nontemporal_store(res[r], out + off);
            }
        }
    }
}

extern "C" void kernel_launch(void* const* d_in, const int* in_sizes, int n_in,
                              void* d_out, int out_size, void* d_ws, size_t ws_size,
                              hipStream_t stream)
{
    // setup_inputs() order: x, W_main, b_main, W_top, b_top, idx_main, idx_top
    const float* x   = (const float*)d_in[0];
    const float* Wm  = (const float*)d_in[1];
    const float* bm  = (const float*)d_in[2];
    const float* Wt  = (const float*)d_in[3];
    const float* bt  = (const float*)d_in[4];
    const int*   Im  = (const int*)  d_in[5];
    const int*   It  = (const int*)  d_in[6];
    float*       out = (float*)d_out;

    const int blocks = NTASK / WPB;               // 112640 / 8 = 14080, exact
    patchy_wmma_kernel<<<blocks, 256, 0, stream>>>(x, Wm, bm, Wt, bt, Im, It, out);
}